// OnlineGCRN_11845519803086
// MI455X (gfx1250) — compile-verified
//
#include <hip/hip_runtime.h>
#include <cmath>
#include <cstdint>

// ---------------------------------------------------------------------------
// OnlineGCRN for MI455X (gfx1250).
//  - All GEMM-shaped work on V_WMMA_F32_16X16X4_F32 (fp32, bit-faithful accum).
//  - Encoder/decoder weight panels staged into LDS with
//    GLOBAL_LOAD_ASYNC_TO_LDS_B128 (ASYNCcnt-tracked, double-buffered), since
//    all 8 waves of a workgroup consume identical B fragments (8x VMEM cut).
//  - GRU GEMM keeps direct coalesced global B loads (per-wave-unique tiles,
//    L2-resident weights), fused r/z/n gate epilogue.
// ---------------------------------------------------------------------------

typedef float v2f __attribute__((ext_vector_type(2)));
typedef float v8f __attribute__((ext_vector_type(8)));

__device__ __forceinline__ v8f wmma_f32_k4(v2f a, v2f b, v8f c) {
  // 8 args: (neg_a, A, neg_b, B, c_mod, C, reuse_a, reuse_b)
  return __builtin_amdgcn_wmma_f32_16x16x4_f32(false, a, false, b, (short)0, c,
                                               false, false);
}

__device__ __forceinline__ float elu1(float v) { return v > 0.0f ? v : (expf(v) - 1.0f); }
__device__ __forceinline__ float sigm(float v) { return 1.0f / (1.0f + expf(-v)); }

__device__ __forceinline__ v8f zero8() {
  v8f z = {0.f, 0.f, 0.f, 0.f, 0.f, 0.f, 0.f, 0.f};
  return z;
}

// Async memory -> LDS copy of one 16B chunk (CDNA5, ASYNCcnt-tracked).
__device__ __forceinline__ void async_copy16(uint32_t lds_off, const float* gaddr) {
  asm volatile("global_load_async_to_lds_b128 %0, %1, off"
               :
               : "v"(lds_off), "v"(gaddr)
               : "memory");
}
__device__ __forceinline__ void async_wait0() {
  asm volatile("s_wait_asynccnt 0x0" ::: "memory");
}

// ---------------------------------------------------------------------------
// Weight-prep kernels (run every call; deterministic).
// ---------------------------------------------------------------------------

// src[N][K] -> dst[K][N]   (GRU: Wih/Whh (2880,960) -> (960,2880))
__global__ __launch_bounds__(256) void transpose_nk(const float* __restrict__ src,
                                                    float* __restrict__ dst,
                                                    int N, int K) {
  size_t i = (size_t)blockIdx.x * blockDim.x + threadIdx.x;
  size_t total = (size_t)N * K;
  if (i >= total) return;
  int n = (int)(i / K);
  int k = (int)(i % K);
  dst[(size_t)k * N + n] = src[i];
}

// Encoder W-cat: dst[(half*CIN+cin)*3+kk][64] = (half? k2 : k1)[cout][cin][kk]
__global__ __launch_bounds__(256) void enc_wprep(const float* __restrict__ k1,
                                                 const float* __restrict__ k2,
                                                 float* __restrict__ dst, int CIN) {
  int i = blockIdx.x * blockDim.x + threadIdx.x;
  int total = 2 * CIN * 3 * 64;
  if (i >= total) return;
  int col = i & 63;
  int row = i >> 6;
  int half = row / (CIN * 3);
  int r = row - half * CIN * 3;
  int cin = r / 3, kk = r - cin * 3;
  const float* s = half ? k2 : k1;  // half0 = prev-frame kernel (k1/gk1)
  dst[i] = s[(col * CIN + cin) * 3 + kk];
}

// Decoder W-cat: dst[kk][half*128+cin][WS] = (half? k2 : k1)[cin][cout][kk], 0-padded
__global__ __launch_bounds__(256) void dec_wprep(const float* __restrict__ k1,
                                                 const float* __restrict__ k2,
                                                 float* __restrict__ dst, int COUT,
                                                 int WS) {
  int i = blockIdx.x * blockDim.x + threadIdx.x;
  int total = 3 * 256 * WS;
  if (i >= total) return;
  int col = i % WS;
  int row = i / WS;
  int kk = row >> 8;
  int rr = row & 255;
  int half = rr >> 7;
  int cin = rr & 127;
  const float* s = half ? k2 : k1;
  dst[i] = (col < COUT) ? s[(cin * COUT + col) * 3 + kk] : 0.0f;
}

// ---------------------------------------------------------------------------
// Encoder gated block: strided conv as GEMM per output position.
//   grid = (Fout, 4), block = 256 (8 waves), wave -> 16 batch rows x 64 cout.
//   Weight panel (shared by all 8 waves) staged via async-to-LDS, dbl-buffered.
// ---------------------------------------------------------------------------
template <int CIN>
__global__ __launch_bounds__(256) void enc_block_kernel(
    const float* __restrict__ x, const float* __restrict__ xprev,
    const float* __restrict__ Wy, const float* __restrict__ Wg,
    const float* __restrict__ by, const float* __restrict__ bg,
    float* __restrict__ out, int Fin, int Fout) {
  constexpr int KT = 2 * CIN * 3;
  constexpr int KC = 16;             // K rows per LDS chunk
  constexpr int NCHUNK = KT / KC;    // 6 (CIN=16) or 24 (CIN=64)
  __shared__ __align__(16) float sW[2][2][KC][64];  // [buf][y/g][k][col] 16KB

  const int tid = threadIdx.x;
  const int t = blockIdx.x;
  const int wave = tid >> 5;
  const int lane = tid & 31;
  const int m0 = (blockIdx.y * 8 + wave) * 16;
  const int lrow = lane & 15;
  const int khalf = lane >> 4;  // A/B frag: lanes 0-15 = K{0,1}, 16-31 = K{2,3}
  const int m = m0 + lrow;

  v8f accY[4], accG[4];
#pragma unroll
  for (int n = 0; n < 4; n++) { accY[n] = zero8(); accG[n] = zero8(); }

  auto stage = [&](int buf, int kbase) {
    // KC*64*4B = 4096B per panel -> one 16B chunk per thread per panel
    uint32_t l0 = (uint32_t)(uintptr_t)(&sW[buf][0][0][0]) + (uint32_t)tid * 16u;
    uint32_t l1 = (uint32_t)(uintptr_t)(&sW[buf][1][0][0]) + (uint32_t)tid * 16u;
    async_copy16(l0, Wy + (size_t)kbase * 64 + (size_t)tid * 4);
    async_copy16(l1, Wg + (size_t)kbase * 64 + (size_t)tid * 4);
  };

  stage(0, 0);
  for (int c = 0; c < NCHUNK; c++) {
    async_wait0();
    __syncthreads();                    // buf c&1 ready in all lanes/waves
    if (c + 1 < NCHUNK) stage((c + 1) & 1, (c + 1) * KC);
    const int buf = c & 1;
#pragma unroll
    for (int ks = 0; ks < KC / 4; ks++) {
      const int kl = ks * 4 + khalf * 2;      // local K row in chunk
      const int kg = c * KC + kl;             // global K row
      v2f a;
#pragma unroll
      for (int e = 0; e < 2; e++) {
        int ki = kg + e;
        int half = (ki >= CIN * 3) ? 1 : 0;
        int r = ki - half * (CIN * 3);
        int cin = r / 3, kk = r - cin * 3;
        const float* s = half ? x : xprev;
        a[e] = s[(size_t)m * CIN * Fin + (size_t)cin * Fin + 2 * t + kk];
      }
#pragma unroll
      for (int n = 0; n < 4; n++) {
        const int col = n * 16 + lrow;
        v2f bY, bG;
        bY[0] = sW[buf][0][kl + 0][col];
        bY[1] = sW[buf][0][kl + 1][col];
        bG[0] = sW[buf][1][kl + 0][col];
        bG[1] = sW[buf][1][kl + 1][col];
        accY[n] = wmma_f32_k4(a, bY, accY[n]);
        accG[n] = wmma_f32_k4(a, bG, accG[n]);
      }
    }
  }

#pragma unroll
  for (int n = 0; n < 4; n++) {
    const int col = n * 16 + lrow;
    const float bYv = by[col], bGv = bg[col];
#pragma unroll
    for (int r = 0; r < 8; r++) {
      const int mm = m0 + khalf * 8 + r;  // C layout: lanes<16 -> M=r, else M=8+r
      float y = accY[n][r] + bYv;
      float g = accG[n][r] + bGv;
      float o = elu1(sigm(g) * y);
      o = elu1(o);
      out[(size_t)mm * 64 * Fout + (size_t)col * Fout + t] = o;
    }
  }
}

// ---------------------------------------------------------------------------
// GRU step: gi = x@Wih^T + bih ; gh = h@Whh^T + bhh ; fused gate epilogue.
//   1920 wave-tiles (32 m-tiles x 60 j-tiles), 6 accumulators per wave.
// ---------------------------------------------------------------------------
__global__ __launch_bounds__(256) void gru_kernel(
    const float* __restrict__ xflat,  // [512][960] (= enc3 output flattened)
    const float* __restrict__ h,      // [512][960]
    const float* __restrict__ WihT,   // [960][2880]
    const float* __restrict__ WhhT,   // [960][2880]
    const float* __restrict__ bih, const float* __restrict__ bhh,
    float* __restrict__ hnew) {       // [512][960]
  const int wave = threadIdx.x >> 5;
  const int lane = threadIdx.x & 31;
  const int tile = blockIdx.x * 8 + wave;  // 0..1919 exactly
  const int jt = tile % 60, mt = tile / 60;
  const int j0 = jt * 16, m0 = mt * 16;
  const int lrow = lane & 15;
  const int khalf = lane >> 4;
  const int m = m0 + lrow;

  v8f acc[6];
#pragma unroll
  for (int i = 0; i < 6; i++) acc[i] = zero8();

  for (int k = 0; k < 960; k += 4) {
    const int k0 = k + khalf * 2;
    v2f ax, ah;
    ax[0] = xflat[(size_t)m * 960 + k0];
    ax[1] = xflat[(size_t)m * 960 + k0 + 1];
    ah[0] = h[(size_t)m * 960 + k0];
    ah[1] = h[(size_t)m * 960 + k0 + 1];
#pragma unroll
    for (int g = 0; g < 3; g++) {
      const int col = g * 960 + j0 + lrow;
      v2f bi, bh;
      bi[0] = WihT[(size_t)(k0 + 0) * 2880 + col];
      bi[1] = WihT[(size_t)(k0 + 1) * 2880 + col];
      bh[0] = WhhT[(size_t)(k0 + 0) * 2880 + col];
      bh[1] = WhhT[(size_t)(k0 + 1) * 2880 + col];
      acc[g]     = wmma_f32_k4(ax, bi, acc[g]);
      acc[3 + g] = wmma_f32_k4(ah, bh, acc[3 + g]);
    }
  }

  const int col = j0 + lrow;
  const float bir = bih[col], biz = bih[960 + col], bin = bih[1920 + col];
  const float bhr = bhh[col], bhz = bhh[960 + col], bhn = bhh[1920 + col];
#pragma unroll
  for (int r = 0; r < 8; r++) {
    const int mm = m0 + khalf * 8 + r;
    const float hv = h[(size_t)mm * 960 + col];
    const float rr = sigm((acc[0][r] + bir) + (acc[3][r] + bhr));
    const float zz = sigm((acc[1][r] + biz) + (acc[4][r] + bhz));
    const float nn = tanhf((acc[2][r] + bin) + rr * (acc[5][r] + bhn));
    hnew[(size_t)mm * 960 + col] = (1.0f - zz) * nn + zz * hv;
  }
}

// ---------------------------------------------------------------------------
// Decoder gated block: conv_transpose1d(stride 2, k=3) as per-output-position
// GEMM over 1-2 valid taps; input channels = [xprev(128) || (xd(64)||enc(64))].
// Weight panels staged per-chunk via async-to-LDS (uniform tap list -> no
// EXEC divergence at WMMA or barriers).
// ---------------------------------------------------------------------------
template <int COUT>
__global__ __launch_bounds__(256) void dec_block_kernel(
    const float* __restrict__ xd,    // [512][64][Fin]
    const float* __restrict__ enc,   // [512][64][Fin]
    const float* __restrict__ xprev, // [512][128][Fin]
    const float* __restrict__ Wy,    // [3][256][WS] (WS = NT*16)
    const float* __restrict__ Wg,
    const float* __restrict__ by, const float* __restrict__ bg,
    float* __restrict__ out, int Fin, int Fout) {  // Fout = 2*Fin+1
  constexpr int NT = (COUT + 15) / 16;
  constexpr int WS = NT * 16;
  constexpr int KC = 16;                  // K rows per LDS chunk
  constexpr int CPT = 256 / KC;           // chunks per tap = 16
  __shared__ __align__(16) float sW[2][2][KC][WS];

  const int tid = threadIdx.x;
  const int t = blockIdx.x;
  const int wave = tid >> 5;
  const int lane = tid & 31;
  const int m0 = (blockIdx.y * 8 + wave) * 16;
  const int lrow = lane & 15;
  const int khalf = lane >> 4;
  const int m = m0 + lrow;

  // Uniform tap list (t is blockIdx.x -> no divergence)
  int fls[3], kks[3], ntap = 0;
  for (int kk = 0; kk < 3; kk++) {
    const int tf = t - kk;
    if (tf >= 0 && !(tf & 1)) {
      const int f = tf >> 1;
      if (f < Fin) { fls[ntap] = f; kks[ntap] = kk; ntap++; }
    }
  }
  const int NCH = ntap * CPT;

  v8f accY[NT], accG[NT];
#pragma unroll
  for (int n = 0; n < NT; n++) { accY[n] = zero8(); accG[n] = zero8(); }

  auto stage = [&](int buf, int cc) {
    if (tid < (KC * WS) / 4) {      // one 16B chunk per issuing thread per panel
      const int tap = cc / CPT;
      const int kb = (cc % CPT) * KC;
      uint32_t l0 = (uint32_t)(uintptr_t)(&sW[buf][0][0][0]) + (uint32_t)tid * 16u;
      uint32_t l1 = (uint32_t)(uintptr_t)(&sW[buf][1][0][0]) + (uint32_t)tid * 16u;
      const size_t gbase = ((size_t)kks[tap] * 256 + kb) * WS + (size_t)tid * 4;
      async_copy16(l0, Wy + gbase);
      async_copy16(l1, Wg + gbase);
    }
  };

  if (NCH > 0) stage(0, 0);
  for (int cc = 0; cc < NCH; cc++) {
    async_wait0();
    __syncthreads();
    if (cc + 1 < NCH) stage((cc + 1) & 1, cc + 1);
    const int buf = cc & 1;
    const int tap = cc / CPT;
    const int kb = (cc % CPT) * KC;
    const int f = fls[tap];
#pragma unroll
    for (int ks = 0; ks < KC / 4; ks++) {
      const int kl = ks * 4 + khalf * 2;
      const int kt = kb + kl;               // K row within 256-deep tap panel
      v2f a;
#pragma unroll
      for (int e = 0; e < 2; e++) {
        const int ki = kt + e;
        const int half = ki >> 7;
        const int cin = ki & 127;
        float v;
        if (half == 0)
          v = xprev[(size_t)m * 128 * Fin + (size_t)cin * Fin + f];
        else if (cin < 64)
          v = xd[(size_t)m * 64 * Fin + (size_t)cin * Fin + f];
        else
          v = enc[(size_t)m * 64 * Fin + (size_t)(cin - 64) * Fin + f];
        a[e] = v;
      }
#pragma unroll
      for (int n = 0; n < NT; n++) {
        const int col = n * 16 + lrow;
        v2f bY, bG;
        bY[0] = sW[buf][0][kl + 0][col];
        bY[1] = sW[buf][0][kl + 1][col];
        bG[0] = sW[buf][1][kl + 0][col];
        bG[1] = sW[buf][1][kl + 1][col];
        accY[n] = wmma_f32_k4(a, bY, accY[n]);
        accG[n] = wmma_f32_k4(a, bG, accG[n]);
      }
    }
  }

#pragma unroll
  for (int n = 0; n < NT; n++) {
    const int col = n * 16 + lrow;
    const float bYv = (col < COUT) ? by[col] : 0.0f;
    const float bGv = (col < COUT) ? bg[col] : 0.0f;
#pragma unroll
    for (int r = 0; r < 8; r++) {
      const int mm = m0 + khalf * 8 + r;
      float y = accY[n][r] + bYv;
      float g = accG[n][r] + bGv;
      float o = elu1(sigm(g) * y);
      o = elu1(o);
      if (col < COUT)
        out[(size_t)mm * COUT * Fout + (size_t)col * Fout + t] = o;
    }
  }
}

// ---------------------------------------------------------------------------
// Host launcher.
//
// d_in ordering assumption: top-level setup_inputs() insertion order, then
// 'params' expanded in JAX pytree order (dict keys sorted):
//   0:x 1-4:xpe0..3 5-8:xpd00..03 9-12:xpd10..13 13:h_gru
//   14+: dec[0][0..3],dec[1][0..3] each {b,gb,gk1,gk2,k1,k2}   (14..61)
//        enc[0..3]   each {b,gb,gk1,gk2,k1,k2}                 (62..85)
//        gru {Whh,Wih,bhh,bih}                                 (86..89)
// d_out: [2*512*255] stacked decoder outputs, then [512*960] h_new.
// ---------------------------------------------------------------------------
extern "C" void kernel_launch(void* const* d_in, const int* in_sizes, int n_in,
                              void* d_out, int out_size, void* d_ws,
                              size_t ws_size, hipStream_t stream) {
  (void)in_sizes; (void)n_in; (void)out_size; (void)ws_size;

  const float* x = (const float*)d_in[0];
  const float* xpe[4];
  for (int i = 0; i < 4; i++) xpe[i] = (const float*)d_in[1 + i];
  const float* xpd[2][4];
  for (int d = 0; d < 2; d++)
    for (int j = 0; j < 4; j++) xpd[d][j] = (const float*)d_in[5 + d * 4 + j];
  const float* h_gru = (const float*)d_in[13];
  auto DEC = [&](int d, int j, int f) { return (const float*)d_in[14 + (d * 4 + j) * 6 + f]; };
  auto ENC = [&](int i, int f) { return (const float*)d_in[62 + i * 6 + f]; };
  const float* Whh = (const float*)d_in[86];
  const float* Wih = (const float*)d_in[87];
  const float* bhh = (const float*)d_in[88];
  const float* bih = (const float*)d_in[89];

  // field indices within a block param dict (sorted): b=0 gb=1 gk1=2 gk2=3 k1=4 k2=5

  // --- workspace layout (floats) ---
  float* ws = (float*)d_ws;
  size_t o = 0;
  float* WihT = ws + o; o += (size_t)960 * 2880;
  float* WhhT = ws + o; o += (size_t)960 * 2880;
  const int ENCCIN[4] = {16, 64, 64, 64};
  float *encWy[4], *encWg[4];
  for (int i = 0; i < 4; i++) {
    const size_t kt = (size_t)2 * ENCCIN[i] * 3 * 64;
    encWy[i] = ws + o; o += kt;
    encWg[i] = ws + o; o += kt;
  }
  float *decWy[2][4], *decWg[2][4];
  for (int d = 0; d < 2; d++)
    for (int j = 0; j < 4; j++) {
      const int WSJ = (j == 3) ? 16 : 64;
      decWy[d][j] = ws + o; o += (size_t)3 * 256 * WSJ;
      decWg[d][j] = ws + o; o += (size_t)3 * 256 * WSJ;
    }
  const int EF[4] = {127, 63, 31, 15};   // encoder output lengths
  const int EFIN[4] = {255, 127, 63, 31};
  float* E[4];
  for (int i = 0; i < 4; i++) { E[i] = ws + o; o += (size_t)512 * 64 * EF[i]; }
  float* DB0 = ws + o; o += (size_t)512 * 64 * 127;
  float* DB1 = ws + o; o += (size_t)512 * 64 * 127;

  // --- weight prep ---
  {
    const int tot = 2880 * 960;
    transpose_nk<<<(tot + 255) / 256, 256, 0, stream>>>(Wih, WihT, 2880, 960);
    transpose_nk<<<(tot + 255) / 256, 256, 0, stream>>>(Whh, WhhT, 2880, 960);
  }
  for (int i = 0; i < 4; i++) {
    const int tot = 2 * ENCCIN[i] * 3 * 64;
    enc_wprep<<<(tot + 255) / 256, 256, 0, stream>>>(ENC(i, 4), ENC(i, 5), encWy[i], ENCCIN[i]);
    enc_wprep<<<(tot + 255) / 256, 256, 0, stream>>>(ENC(i, 2), ENC(i, 3), encWg[i], ENCCIN[i]);
  }
  for (int d = 0; d < 2; d++)
    for (int j = 0; j < 4; j++) {
      const int COUT = (j == 3) ? 1 : 64;
      const int WSJ = (j == 3) ? 16 : 64;
      const int tot = 3 * 256 * WSJ;
      dec_wprep<<<(tot + 255) / 256, 256, 0, stream>>>(DEC(d, j, 4), DEC(d, j, 5), decWy[d][j], COUT, WSJ);
      dec_wprep<<<(tot + 255) / 256, 256, 0, stream>>>(DEC(d, j, 2), DEC(d, j, 3), decWg[d][j], COUT, WSJ);
    }

  // --- encoder ---
  enc_block_kernel<16><<<dim3(EF[0], 4), 256, 0, stream>>>(
      x, xpe[0], encWy[0], encWg[0], ENC(0, 0), ENC(0, 1), E[0], EFIN[0], EF[0]);
  for (int i = 1; i < 4; i++) {
    enc_block_kernel<64><<<dim3(EF[i], 4), 256, 0, stream>>>(
        E[i - 1], xpe[i], encWy[i], encWg[i], ENC(i, 0), ENC(i, 1), E[i], EFIN[i], EF[i]);
  }

  // --- GRU ---
  float* out = (float*)d_out;
  float* hnew = out + (size_t)2 * 512 * 255;
  gru_kernel<<<240, 256, 0, stream>>>(E[3], h_gru, WihT, WhhT, bih, bhh, hnew);

  // --- decoders ---
  for (int d = 0; d < 2; d++) {
    dec_block_kernel<64><<<dim3(31, 4), 256, 0, stream>>>(
        hnew, E[3], xpd[d][0], decWy[d][0], decWg[d][0], DEC(d, 0, 0), DEC(d, 0, 1),
        DB0, 15, 31);
    dec_block_kernel<64><<<dim3(63, 4), 256, 0, stream>>>(
        DB0, E[2], xpd[d][1], decWy[d][1], decWg[d][1], DEC(d, 1, 0), DEC(d, 1, 1),
        DB1, 31, 63);
    dec_block_kernel<64><<<dim3(127, 4), 256, 0, stream>>>(
        DB1, E[1], xpd[d][2], decWy[d][2], decWg[d][2], DEC(d, 2, 0), DEC(d, 2, 1),
        DB0, 63, 127);
    dec_block_kernel<1><<<dim3(255, 4), 256, 0, stream>>>(
        DB0, E[0], xpd[d][3], decWy[d][3], decWg[d][3], DEC(d, 3, 0), DEC(d, 3, 1),
        out + (size_t)d * 512 * 255, 127, 255);
  }
}